// QCNN_ZNOTY_12756052869700
// MI455X (gfx1250) — compile-verified
//
#include <hip/hip_runtime.h>
#include <hip/hip_bf16.h>
#include <math.h>

typedef __attribute__((ext_vector_type(2))) float v2f;
typedef __attribute__((ext_vector_type(8))) float v8f;

#define PI_F 3.14159265358979323846f
#define NQ 8
#define DIM 256   // 2^8 amplitudes

// ---------------------------------------------------------------------------
// Kernel 1: build the 256x256 circuit unitary, one basis column per block.
// State (256 complex amplitudes) lives in LDS; gates partition indices among
// the 128 threads, so only an end-of-gate barrier is needed.
// Qubit w <-> bit (7-w) of the flattened index (reference reshape order).
// ---------------------------------------------------------------------------
__global__ __launch_bounds__(128) void qcnn_build_unitary(
    const float* __restrict__ conv_rz, const float* __restrict__ conv_ry,
    const float* __restrict__ conv_ry2, const float* __restrict__ pool,
    float* __restrict__ Ure, float* __restrict__ Uim)
{
    __shared__ float sre[DIM];
    __shared__ float sim[DIM];
    const int tid = threadIdx.x;
    const int col = blockIdx.x;

    for (int i = tid; i < DIM; i += 128) { sre[i] = (i == col) ? 1.f : 0.f; sim[i] = 0.f; }
    __syncthreads();

    auto rz = [&](int w, float t) {
        float cp = cosf(0.5f * t), sp = sinf(0.5f * t);
        int b = 7 - w;
        for (int i = tid; i < DIM; i += 128) {
            float ph = ((i >> b) & 1) ? sp : -sp;   // e^{+-i t/2}
            float re = sre[i], im = sim[i];
            sre[i] = re * cp - im * ph;
            sim[i] = re * ph + im * cp;
        }
        __syncthreads();
    };
    auto ry = [&](int w, float t) {
        float c = cosf(0.5f * t), s = sinf(0.5f * t);
        int b = 7 - w;
        int p  = tid;                                   // 128 pairs
        int i0 = ((p >> b) << (b + 1)) | (p & ((1 << b) - 1));
        int i1 = i0 | (1 << b);
        float r0 = sre[i0], m0 = sim[i0], r1 = sre[i1], m1 = sim[i1];
        sre[i0] = c * r0 - s * r1;  sim[i0] = c * m0 - s * m1;
        sre[i1] = s * r0 + c * r1;  sim[i1] = s * m0 + c * m1;
        __syncthreads();
    };
    auto u3g = [&](int w, float th, float ph, float la) {
        float ct = cosf(0.5f * th), st = sinf(0.5f * th);
        float m01r = -cosf(la) * st,      m01i = -sinf(la) * st;
        float m10r =  cosf(ph) * st,      m10i =  sinf(ph) * st;
        float m11r =  cosf(ph + la) * ct, m11i =  sinf(ph + la) * ct;
        int b = 7 - w;
        int p  = tid;
        int i0 = ((p >> b) << (b + 1)) | (p & ((1 << b) - 1));
        int i1 = i0 | (1 << b);
        float r0 = sre[i0], m0 = sim[i0], r1 = sre[i1], m1 = sim[i1];
        sre[i0] = ct * r0 + m01r * r1 - m01i * m1;
        sim[i0] = ct * m0 + m01r * m1 + m01i * r1;
        sre[i1] = m10r * r0 - m10i * m0 + m11r * r1 - m11i * m1;
        sim[i1] = m10r * m0 + m10i * r0 + m11r * m1 + m11i * r1;
        __syncthreads();
    };
    auto cx = [&](int c, int t) {                       // control c, target t
        if (tid < 64) {
            int bc = 7 - c, bt = 7 - t;
            int blo = bc < bt ? bc : bt;
            int bhi = bc < bt ? bt : bc;
            int x = tid;
            x = ((x >> blo) << (blo + 1)) | (x & ((1 << blo) - 1));
            x = ((x >> bhi) << (bhi + 1)) | (x & ((1 << bhi) - 1));
            int i0 = x | (1 << bc);                     // control=1, target=0
            int i1 = i0 | (1 << bt);
            float tr = sre[i0]; sre[i0] = sre[i1]; sre[i1] = tr;
            float ti = sim[i0]; sim[i0] = sim[i1]; sim[i1] = ti;
        }
        __syncthreads();
    };

    const int pq0[10] = {0, 2, 4, 6, 1, 3, 5, 1, 5, 3};
    const int pq1[10] = {1, 3, 5, 7, 2, 4, 6, 3, 7, 5};
    const int u3w0[4] = {1, 3, 5, 7};

    for (int pp = 0; pp < 10; ++pp) {
        int q0 = pq0[pp], q1 = pq1[pp];
        rz(q1, -0.5f * PI_F);
        cx(q1, q0);
        rz(q0, conv_rz[pp]);
        ry(q1, conv_ry[pp]);
        cx(q0, q1);
        ry(q1, conv_ry2[pp]);
        cx(q1, q0);
        rz(q0, 0.5f * PI_F);
        if (pp == 6) {                                  // end of layer 0: pool U3s
            for (int k = 0; k < 4; ++k)
                u3g(u3w0[k], pool[3 * k + 0], pool[3 * k + 1], pool[3 * k + 2]);
        }
    }
    u3g(3, pool[12], pool[13], pool[14]);               // layer 1 pool
    u3g(7, pool[15], pool[16], pool[17]);

    for (int i = tid; i < DIM; i += 128) {              // column col of U, row-major
        Ure[i * DIM + col] = sre[i];
        Uim[i * DIM + col] = sim[i];
    }
}

// ---------------------------------------------------------------------------
// Kernel 2: fused  Psi(x) -> f32 WMMA GEMMs (re/im) -> sign-weighted |c|^2
// reduction -> MLP -> sigmoid.
// One wave per block; each wave owns TWO 16-row M-tiles (32 batch rows) so
// every B (unitary) fragment load feeds 4 WMMAs instead of 2.
// ---------------------------------------------------------------------------
__global__ __launch_bounds__(32) void qcnn_main(
    const float* __restrict__ x,
    const float* __restrict__ W1, const float* __restrict__ b1,
    const float* __restrict__ W2, const float* __restrict__ b2,
    const float* __restrict__ Ure, const float* __restrict__ Uim,
    float* __restrict__ out, int Btot)
{
    __shared__ float psi[32][DIM];                      // 32 KB
    __shared__ float feats[32][2];

    const int lane = threadIdx.x & 31;
    const int rowBlock = blockIdx.x * 32;

    // ---- build Psi rows: lane builds rows (lane&15) and (lane&15)+16,
    //      half of the 256 state indices each (lane>>4 picks the half).
    const int ibase = (lane >> 4) * 128;
#pragma unroll
    for (int half = 0; half < 2; ++half) {
        const int r = half * 16 + (lane & 15);          // local row 0..31
        int grow = rowBlock + r;
        if (grow >= Btot) grow = Btot - 1;
        float c8[NQ], s8[NQ];
#pragma unroll
        for (int w = 0; w < NQ; ++w) {
            float a = 0.5f * x[grow * NQ + w];
            sincosf(a, &s8[w], &c8[w]);
        }
        for (int t = 0; t < 128; ++t) {
            int i = ibase + t;
            float p = 1.f;
#pragma unroll
            for (int b = 0; b < NQ; ++b)
                p *= ((i >> b) & 1) ? s8[7 - b] : c8[7 - b];
            psi[r][i] = p;
        }
    }
    __syncthreads();                                    // single-wave: cheap

    // ---- WMMA GEMM + epilogue
    const int n_loc = lane & 15;                        // column within 16-tile
    const int kb    = (lane >> 4) << 1;                 // K sub-offset 0 or 2
    const int ra    = lane & 15;                        // A row (per tile)
    float acc[4][8];                                    // [featA/B x tile0/1][8 rows]
#pragma unroll
    for (int q = 0; q < 4; ++q)
#pragma unroll
        for (int v = 0; v < 8; ++v) acc[q][v] = 0.f;

    for (int ct = 0; ct < 16; ++ct) {
        v8f dre0 = {0.f,0.f,0.f,0.f,0.f,0.f,0.f,0.f};
        v8f dim0 = {0.f,0.f,0.f,0.f,0.f,0.f,0.f,0.f};
        v8f dre1 = {0.f,0.f,0.f,0.f,0.f,0.f,0.f,0.f};
        v8f dim1 = {0.f,0.f,0.f,0.f,0.f,0.f,0.f,0.f};
        const float* bre = Ure + (ct * 16 + n_loc) * DIM + kb;
        const float* bim = Uim + (ct * 16 + n_loc) * DIM + kb;
        if (ct < 15) {                                  // prefetch next U tile rows
            __builtin_prefetch(bre + DIM * 16, 0, 3);
            __builtin_prefetch(bim + DIM * 16, 0, 3);
        }
#pragma unroll 4
        for (int ks = 0; ks < 64; ++ks) {
            v2f a0 = *(const v2f*)&psi[ra][4 * ks + kb];
            v2f a1 = *(const v2f*)&psi[ra + 16][4 * ks + kb];
            v2f br = *(const v2f*)(bre + 4 * ks);
            v2f bi = *(const v2f*)(bim + 4 * ks);
            dre0 = __builtin_amdgcn_wmma_f32_16x16x4_f32(
                false, a0, false, br, (short)0, dre0, false, false);
            dim0 = __builtin_amdgcn_wmma_f32_16x16x4_f32(
                false, a0, false, bi, (short)0, dim0, false, false);
            dre1 = __builtin_amdgcn_wmma_f32_16x16x4_f32(
                false, a1, false, br, (short)0, dre1, false, false);
            dim1 = __builtin_amdgcn_wmma_f32_16x16x4_f32(
                false, a1, false, bi, (short)0, dim1, false, false);
        }
        const int n_g = ct * 16 + n_loc;                // global state index (column)
        const float sA = (n_g & 16) ? -1.f : 1.f;       // wire 3 <-> bit 4
        const float sB = (n_g & 1)  ? -1.f : 1.f;       // wire 7 <-> bit 0
#pragma unroll
        for (int v = 0; v < 8; ++v) {
            float m0 = dre0[v] * dre0[v] + dim0[v] * dim0[v];
            float m1 = dre1[v] * dre1[v] + dim1[v] * dim1[v];
            acc[0][v] += sA * m0;
            acc[1][v] += sB * m0;
            acc[2][v] += sA * m1;
            acc[3][v] += sB * m1;
        }
    }

    // reduce across the 16 lanes of each half-wave (rows 0-7 | rows 8-15)
#pragma unroll
    for (int m = 1; m < 16; m <<= 1)
#pragma unroll
        for (int q = 0; q < 4; ++q)
#pragma unroll
            for (int v = 0; v < 8; ++v)
                acc[q][v] += __shfl_xor(acc[q][v], m);

    if ((lane & 15) == 0) {
        int base = (lane >> 4) << 3;                    // 0 for lanes 0-15, 8 for 16-31
#pragma unroll
        for (int v = 0; v < 8; ++v) {
            feats[base + v][0]      = acc[0][v];        // tile0 rows 0..15
            feats[base + v][1]      = acc[1][v];
            feats[base + v + 16][0] = acc[2][v];        // tile1 rows 16..31
            feats[base + v + 16][1] = acc[3][v];
        }
    }
    __syncthreads();

    // ---- fused MLP head: tanh(feats @ W1^T + b1) @ W2^T + b2 -> sigmoid
    {
        int orow = rowBlock + lane;
        if (orow < Btot) {
            float f0 = feats[lane][0];
            float f1 = feats[lane][1];
            float o = b2[0];
#pragma unroll
            for (int j = 0; j < 10; ++j)
                o += W2[j] * tanhf(W1[2 * j] * f0 + W1[2 * j + 1] * f1 + b1[j]);
            out[orow] = 1.f / (1.f + expf(-o));
        }
    }
}

// ---------------------------------------------------------------------------
extern "C" void kernel_launch(void* const* d_in, const int* in_sizes, int n_in,
                              void* d_out, int out_size, void* d_ws, size_t ws_size,
                              hipStream_t stream) {
    const float* x        = (const float*)d_in[0];
    const float* conv_rz  = (const float*)d_in[1];
    const float* conv_ry  = (const float*)d_in[2];
    const float* conv_ry2 = (const float*)d_in[3];
    const float* pool     = (const float*)d_in[4];
    const float* W1       = (const float*)d_in[5];
    const float* b1       = (const float*)d_in[6];
    const float* W2       = (const float*)d_in[7];
    const float* b2       = (const float*)d_in[8];

    float* Ure = (float*)d_ws;                // 256*256 floats
    float* Uim = Ure + DIM * DIM;             // 256*256 floats (512 KB total)

    const int B = in_sizes[0] / NQ;

    qcnn_build_unitary<<<DIM, 128, 0, stream>>>(conv_rz, conv_ry, conv_ry2, pool,
                                                Ure, Uim);
    qcnn_main<<<(B + 31) / 32, 32, 0, stream>>>(x, W1, b1, W2, b2, Ure, Uim,
                                                (float*)d_out, B);
}